// HSTUPositionalEncoder_67680094650961
// MI455X (gfx1250) — compile-verified
//
#include <hip/hip_runtime.h>
#include <stdint.h>

// ---------------------------------------------------------------------------
// HSTU positional encoder: out = emb * sqrt(D) + pos_weight[pos(tok)]
// Memory-bound (0.17 FLOP/B, ~272 MB -> ~11.7 us floor at 23.3 TB/s).
// CDNA5 path: async global->LDS staging of the gathered pos_weight rows
// (ASYNCcnt pipeline, double buffered), NT hints on streaming emb/out,
// wave32-private LDS slots (no barriers in the hot loop).
// ---------------------------------------------------------------------------

typedef __attribute__((ext_vector_type(4))) float f4;
typedef __attribute__((ext_vector_type(4))) int   v4i;

typedef __attribute__((address_space(1))) v4i g128_t;   // global (AS1)
typedef __attribute__((address_space(3))) v4i l128_t;   // LDS (AS3)

constexpr int DIM     = 512;   // embedding dim (reference: D=512)
constexpr int THREADS = 256;   // 8 waves (wave32)
constexpr int TPB     = 2;     // tokens per block-iteration: 256*4/512
constexpr int MAXB    = 128;   // max sequences cached in LDS (B=16 actual)

__device__ __forceinline__ void async_ld_b128(const float* gsrc, uint32_t lds_byte_addr) {
#if __has_builtin(__builtin_amdgcn_global_load_async_to_lds_b128)
  __builtin_amdgcn_global_load_async_to_lds_b128(
      (g128_t*)(uintptr_t)gsrc, (l128_t*)lds_byte_addr, /*offset=*/0, /*cpol=*/0);
#else
  asm volatile("global_load_async_to_lds_b128 %0, %1, off"
               :: "v"(lds_byte_addr), "v"((uint64_t)(uintptr_t)gsrc)
               : "memory");
#endif
}

__global__ __launch_bounds__(THREADS) void hstu_posenc_kernel(
    const int*   __restrict__ seq_lengths,   // [B]
    const int*   __restrict__ seq_offsets,   // [B+1]
    const float* __restrict__ emb,           // [total, DIM]
    const float* __restrict__ pw,            // [P, DIM]
    float*       __restrict__ out,           // [total, DIM]
    int nB, int total, int maxPos)
{
  __shared__ int s_off[MAXB + 1];
  __shared__ int s_high[MAXB];
  __shared__ f4  s_pw[2][THREADS];           // 8 KB double buffer

  const int t = threadIdx.x;
  if (t <= nB && t <= MAXB) s_off[t] = seq_offsets[t];
  if (t <  nB && t <  MAXB) {
    int sl = seq_lengths[t];
    int pm = maxPos - 1;
    s_high[t] = sl < pm ? sl : pm;
  }
  __syncthreads();

  const int   sub   = t >> 7;                // which of the 2 tokens
  const int   col   = (t & 127) << 2;        // float column (16B aligned)
  const float alpha = 22.62741699796952f;    // sqrt(512)
  const uint32_t lds_base = (uint32_t)(uintptr_t)&s_pw[0][0]; // low32 == LDS addr

  // pos(tok): searchsorted-right segment scan over B+1 offsets, then clamp.
  auto pos_of = [&](int tok) -> int {
    int seg = 0;
    #pragma unroll 4
    for (int j = 1; j < nB; ++j) seg += (s_off[j] <= tok) ? 1 : 0;
    int p = s_high[seg] - (tok - s_off[seg]);
    p = p < 0 ? 0 : p;
    int pm = maxPos - 1;
    return p > pm ? pm : p;
  };

  // Stage this thread's 16B of the gathered pos_weight row into its own slot.
  auto stage = [&](int base, int buf) {
    int tok = base + sub;
    if (tok >= total) tok = total - 1;       // harmless duplicate on tail
    int p = pos_of(tok);
    const float* src = pw + (size_t)p * DIM + col;
    uint32_t laddr = lds_base + (uint32_t)((buf * THREADS + t) * sizeof(f4));
    async_ld_b128(src, laddr);
  };

  const int stride = (int)gridDim.x * TPB;
  int base = (int)blockIdx.x * TPB;
  if (base >= total) return;

  stage(base, 0);                            // prologue: prefetch iter 0
  int buf = 0;
  while (base < total) {
    int  nbase   = base + stride;
    bool hasnext = nbase < total;            // uniform per block
    if (hasnext) {
      stage(nbase, buf ^ 1);                 // prefetch next iteration
      asm volatile("s_wait_asynccnt 1" ::: "memory");  // current buffer landed
    } else {
      asm volatile("s_wait_asynccnt 0" ::: "memory");  // drain
    }

    int tok = base + sub;
    if (tok < total) {
      f4 pv = s_pw[buf][t];                                            // ds_load_b128
      f4 ev = __builtin_nontemporal_load(
                  (const f4*)(emb + (size_t)tok * DIM + col));         // NT stream in
      f4 r;
      r.x = __builtin_fmaf(ev.x, alpha, pv.x);
      r.y = __builtin_fmaf(ev.y, alpha, pv.y);
      r.z = __builtin_fmaf(ev.z, alpha, pv.z);
      r.w = __builtin_fmaf(ev.w, alpha, pv.w);
      __builtin_nontemporal_store(r, (f4*)(out + (size_t)tok * DIM + col)); // NT stream out
    }
    buf ^= 1;
    base = nbase;
  }
}

extern "C" void kernel_launch(void* const* d_in, const int* in_sizes, int n_in,
                              void* d_out, int out_size, void* d_ws, size_t ws_size,
                              hipStream_t stream) {
  // setup_inputs order: max_seq_len(1), seq_lengths[B], seq_offsets[B+1],
  //                     seq_embeddings[total*D], pos_weight[P*D]
  const int*   seq_lengths = (const int*)d_in[1];
  const int*   seq_offsets = (const int*)d_in[2];
  const float* emb         = (const float*)d_in[3];
  const float* pw          = (const float*)d_in[4];
  float*       out         = (float*)d_out;

  const int nB     = in_sizes[1];
  const int total  = in_sizes[3] / DIM;
  const int maxPos = in_sizes[4] / DIM;

  int npairs = (total + TPB - 1) / TPB;
  int grid   = npairs < 4096 ? npairs : 4096;   // ~8 pipelined iters per block

  hipLaunchKernelGGL(hstu_posenc_kernel, dim3(grid), dim3(THREADS), 0, stream,
                     seq_lengths, seq_offsets, emb, pw, out, nB, total, maxPos);
}